// RelGraphConvLayer_4861902979422
// MI455X (gfx1250) — compile-verified
//
#include <hip/hip_runtime.h>

#define N_NODES 100000
#define N_RELS  4
#define N_EDGES 200000
#define D_IN    256
#define D_OUT   256
#define LN_EPS  1e-5f

typedef float v2f __attribute__((ext_vector_type(2)));
typedef float v4f __attribute__((ext_vector_type(4)));
typedef float v8f __attribute__((ext_vector_type(8)));

// ---------------------------------------------------------------------------
// Kernel 1: per-relation in-degree (float, so division is direct later)
// ---------------------------------------------------------------------------
__global__ void rgcn_degree_kernel(const int* __restrict__ dst_idx,
                                   float* __restrict__ deg) {
    int e = blockIdx.x * blockDim.x + threadIdx.x;
    if (e >= N_RELS * N_EDGES) return;
    int r = e / N_EDGES;
    int dst = dst_idx[e];
    atomicAdd(&deg[r * N_NODES + dst], 1.0f);
}

// ---------------------------------------------------------------------------
// Kernel 2: fused scatter. For each edge (r, src->dst):
//   Z_b[dst] += (w_comp[r,b] / max(deg_r[dst],1)) * x[src]   for b = 0,1
// One wave (32 lanes) per edge; each lane moves 8 of the 256 features.
// ---------------------------------------------------------------------------
__global__ __launch_bounds__(256)
void rgcn_scatter_kernel(const float* __restrict__ x,
                         const int* __restrict__ src_idx,
                         const int* __restrict__ dst_idx,
                         const float* __restrict__ w_comp,
                         const float* __restrict__ deg,
                         float* __restrict__ Z0,
                         float* __restrict__ Z1) {
    const int lane = threadIdx.x & 31;
    const int e = blockIdx.x * (blockDim.x >> 5) + (threadIdx.x >> 5);
    if (e >= N_RELS * N_EDGES) return;

    const int r   = e / N_EDGES;
    const int src = src_idx[e];
    const int dst = dst_idx[e];
    const float inv = 1.0f / fmaxf(deg[r * N_NODES + dst], 1.0f);
    const float s0 = w_comp[r * 2 + 0] * inv;
    const float s1 = w_comp[r * 2 + 1] * inv;

    const float* xr = x + (size_t)src * D_IN + lane * 8;
    float* z0 = Z0 + (size_t)dst * D_IN + lane * 8;
    float* z1 = Z1 + (size_t)dst * D_IN + lane * 8;

    v4f a = *(const v4f*)(xr);
    v4f b = *(const v4f*)(xr + 4);
#pragma unroll
    for (int i = 0; i < 4; ++i) {
        atomicAdd(&z0[i], s0 * a[i]);
        atomicAdd(&z1[i], s1 * a[i]);
    }
#pragma unroll
    for (int i = 0; i < 4; ++i) {
        atomicAdd(&z0[4 + i], s0 * b[i]);
        atomicAdd(&z1[4 + i], s1 * b[i]);
    }
}

// ---------------------------------------------------------------------------
// Kernel 3: fused GEMM + bias + LayerNorm + ReLU.
//   out[16 rows x 256 cols per block] =
//     LN( [Z0|Z1|x](16x768) @ [bases0;bases1;loop_w](768x256) + bias )
// 8 waves/block; wave w owns cols [32w, 32w+32) via two f32 WMMA accumulators.
// A panel staged in LDS (stride 772 to avoid bank conflicts), B streamed
// from L2-resident weights. WMMA: V_WMMA_F32_16X16X4_F32.
// ---------------------------------------------------------------------------
__global__ __launch_bounds__(256)
void rgcn_gemm_ln_kernel(const float* __restrict__ Z0,
                         const float* __restrict__ Z1,
                         const float* __restrict__ x,
                         const float* __restrict__ bases,
                         const float* __restrict__ loopw,
                         const float* __restrict__ h_bias,
                         const float* __restrict__ gamma,
                         const float* __restrict__ beta,
                         float* __restrict__ out) {
    __shared__ __align__(16) float smem[16 * 772];   // A tile; reused as h tile
    __shared__ float red_s[16 * 16];
    __shared__ float red_q[16 * 16];
    __shared__ float s_mu[16];
    __shared__ float s_rs[16];

    const int tid = threadIdx.x;
    const int rowBase = blockIdx.x * 16;

    // ---- stage A = [Z0 | Z1 | x] rows into LDS: 16 x 768, padded stride 772
    for (int i = tid; i < (16 * 768) / 4; i += 256) {
        int flat = i * 4;
        int row = flat / 768;
        int col = flat - row * 768;
        int seg = col >> 8;
        int c = col & 255;
        size_t off = (size_t)(rowBase + row) * D_IN + c;
        const float* p = (seg == 0) ? (Z0 + off) : (seg == 1) ? (Z1 + off) : (x + off);
        *(v4f*)&smem[row * 772 + col] = *(const v4f*)p;
    }
    __syncthreads();

    const int lane = tid & 31;
    const int wave = tid >> 5;
    const int colBase = wave * 32;
    const int arow = lane & 15;          // M row / N col within tile
    const int kh = (lane >> 4) * 2;      // lanes 16-31 hold K=2,3

    v8f c0 = {};
    v8f c1 = {};
#pragma unroll
    for (int seg = 0; seg < 3; ++seg) {
        const float* W = (seg == 0) ? bases
                       : (seg == 1) ? (bases + 256 * 256)
                                    : loopw;
        const float* arowp = &smem[arow * 772 + seg * 256];
        for (int k = 0; k < 256; k += 4) {
            const int krow = k + kh;
            v2f a = *(const v2f*)(arowp + krow);           // ds_load_b64
            const float* wp = W + krow * 256 + colBase + arow;
            v2f b0; b0.x = wp[0];  b0.y = wp[256];
            c0 = __builtin_amdgcn_wmma_f32_16x16x4_f32(
                     false, a, false, b0, (short)0, c0, false, false);
            v2f b1; b1.x = wp[16]; b1.y = wp[272];
            c1 = __builtin_amdgcn_wmma_f32_16x16x4_f32(
                     false, a, false, b1, (short)0, c1, false, false);
        }
    }
    __syncthreads();   // done reading A tile -> reuse smem for h tile

    // ---- spill accumulators to LDS h tile [16][260]
    {
        const int hrow0 = (lane >> 4) * 8;       // C/D layout: lanes>=16 -> M+8
        const int hcol = colBase + (lane & 15);
#pragma unroll
        for (int v = 0; v < 8; ++v) {
            smem[(hrow0 + v) * 260 + hcol]      = c0[v];
            smem[(hrow0 + v) * 260 + hcol + 16] = c1[v];
        }
    }
    __syncthreads();

    // ---- LayerNorm: partial sums (thread t: row t/16, 16-col segment t%16)
    const int prow = tid >> 4;
    const int pseg = tid & 15;
    float sum = 0.f, sq = 0.f;
#pragma unroll
    for (int i = 0; i < 16; ++i) {
        int col = pseg * 16 + i;
        float h = smem[prow * 260 + col] + h_bias[col];
        sum += h;
        sq += h * h;
    }
    red_s[prow * 16 + pseg] = sum;
    red_q[prow * 16 + pseg] = sq;
    __syncthreads();

    if (tid < 16) {
        float s = 0.f, q = 0.f;
#pragma unroll
        for (int i = 0; i < 16; ++i) {
            s += red_s[tid * 16 + i];
            q += red_q[tid * 16 + i];
        }
        float mu = s * (1.0f / 256.0f);
        float var = q * (1.0f / 256.0f) - mu * mu;
        s_mu[tid] = mu;
        s_rs[tid] = rsqrtf(var + LN_EPS);
    }
    __syncthreads();

    // ---- normalize + affine + ReLU, b128 stores
    {
        const float mu = s_mu[prow];
        const float rs = s_rs[prow];
        float* op = out + (size_t)(rowBase + prow) * D_OUT + pseg * 16;
#pragma unroll
        for (int g = 0; g < 4; ++g) {
            v4f o;
#pragma unroll
            for (int i = 0; i < 4; ++i) {
                int col = pseg * 16 + g * 4 + i;
                float h = smem[prow * 260 + col] + h_bias[col];
                h = (h - mu) * rs * gamma[col] + beta[col];
                o[i] = fmaxf(h, 0.f);
            }
            *(v4f*)(op + g * 4) = o;
        }
    }
}

// ---------------------------------------------------------------------------
extern "C" void kernel_launch(void* const* d_in, const int* in_sizes, int n_in,
                              void* d_out, int out_size, void* d_ws, size_t ws_size,
                              hipStream_t stream) {
    const float* x        = (const float*)d_in[0];
    const int*   src_idx  = (const int*)d_in[1];
    const int*   dst_idx  = (const int*)d_in[2];
    const float* bases    = (const float*)d_in[3];
    const float* w_comp   = (const float*)d_in[4];
    const float* loopw    = (const float*)d_in[5];
    const float* h_bias   = (const float*)d_in[6];
    const float* ln_gamma = (const float*)d_in[7];
    const float* ln_beta  = (const float*)d_in[8];
    float* out = (float*)d_out;

    // workspace layout: Z0 [N, 256] | Z1 [N, 256] | deg [R, N]
    float* Z0  = (float*)d_ws;
    float* Z1  = Z0 + (size_t)N_NODES * D_IN;
    float* deg = Z1 + (size_t)N_NODES * D_IN;
    size_t zero_bytes = ((size_t)N_NODES * D_IN * 2 + (size_t)N_RELS * N_NODES)
                        * sizeof(float);
    hipMemsetAsync(d_ws, 0, zero_bytes, stream);

    const int tot_edges = N_RELS * N_EDGES;
    rgcn_degree_kernel<<<(tot_edges + 255) / 256, 256, 0, stream>>>(dst_idx, deg);

    // one wave per edge, 8 edges per 256-thread block
    rgcn_scatter_kernel<<<tot_edges / 8, 256, 0, stream>>>(
        x, src_idx, dst_idx, w_comp, deg, Z0, Z1);

    // 16 rows per block, 100000 / 16 = 6250 blocks
    rgcn_gemm_ln_kernel<<<N_NODES / 16, 256, 0, stream>>>(
        Z0, Z1, x, bases, loopw, h_bias, ln_gamma, ln_beta, out);
}